// RoIAwarePool3d_23845658427780
// MI455X (gfx1250) — compile-verified
//
#include <hip/hip_runtime.h>
#include <math.h>

// RoIAware 3D pooling for MI455X (gfx1250).
// CDNA5 paths: global_load_async_to_lds_b128/b32 (async DMA, ASYNCcnt),
// s_wait_asynccnt, wave32 workgroup barriers, global u32 atomics.
//
// Occupancy-tuned: grid = (P/1024 point tiles) x (N/16 roi groups) ~= 392
// blocks of 256 threads. Async DMA of the coord tile is issued first and
// overlapped with per-roi constant derivation (cos/sin), then one wait +
// one barrier, then register-resident points sweep the LDS roi table.

#define OUTG 12
#define VOXN (OUTG * OUTG * OUTG)   // 1728
#define CCH  16
#define TPB 256
#define CHUNK_PTS 1024              // points staged per block (12 KB LDS)
#define ROIS_PER_BLOCK 16
#define PTS_PER_THREAD (CHUNK_PTS / TPB)   // 4

// ---- order-preserving f32 <-> u32 encoding (monotonic; atomicMax-safe) ----
__device__ __forceinline__ unsigned enc_f32(float f) {
  unsigned u = __float_as_uint(f);
  return (u & 0x80000000u) ? ~u : (u | 0x80000000u);
}
__device__ __forceinline__ float dec_f32(unsigned e) {
  return (e & 0x80000000u) ? __uint_as_float(e & 0x7FFFFFFFu)
                           : __uint_as_float(~e);
}
#define ENC_NEGINF 0x007FFFFFu      // enc(-inf): empty-voxel sentinel

// ---- CDNA5 async global->LDS DMA (inline asm; portable across toolchains) --
#define ASYNC_LDS_B128(ldsoff, gaddr)                                   \
  asm volatile("global_load_async_to_lds_b128 %0, %1, off"              \
               :: "v"(ldsoff), "v"(gaddr) : "memory")
#define ASYNC_LDS_B32(ldsoff, gaddr)                                    \
  asm volatile("global_load_async_to_lds_b32 %0, %1, off"               \
               :: "v"(ldsoff), "v"(gaddr) : "memory")
#define WAIT_ASYNC0() asm volatile("s_wait_asynccnt 0" ::: "memory")

// gfx1250 flat aperture maps LDS as addr[31:0] == LDS byte offset, so
// truncating the generic shared pointer yields the async-DMA VDST operand.
__device__ __forceinline__ unsigned lds_off32(const void* p) {
  return (unsigned)(size_t)p;
}

// ---------------------------------------------------------------------------
__global__ void roiaware_init(unsigned* __restrict__ out_u,
                              unsigned* __restrict__ cnt,
                              const int* __restrict__ mode_p,
                              int total, int nvox, int has_cnt) {
  const int mode = mode_p[0];
  const unsigned fill = (mode == 0) ? ENC_NEGINF : 0u;
  for (int i = blockIdx.x * blockDim.x + threadIdx.x; i < total;
       i += gridDim.x * blockDim.x) {
    out_u[i] = fill;
    if (has_cnt && i < nvox) cnt[i] = 0u;
  }
}

// ---------------------------------------------------------------------------
__global__ __launch_bounds__(TPB)
void roiaware_pool(const float* __restrict__ rois,
                   const float* __restrict__ pts,
                   const float* __restrict__ feat,
                   const int* __restrict__ mode_p,
                   unsigned* __restrict__ out_u,
                   float* __restrict__ out_f,
                   unsigned* __restrict__ cnt,
                   int N, int P, int has_cnt) {
  __shared__ float sp[CHUNK_PTS * 3];          // staged point coords (x,y,z)
  __shared__ float sroi[ROIS_PER_BLOCK][12];   // derived roi params

  const int tid     = threadIdx.x;
  const int mode    = mode_p[0];
  const int cstart  = blockIdx.x * CHUNK_PTS;
  const int roiBase = blockIdx.y * ROIS_PER_BLOCK;
  const int nroiLoc = (N - roiBase < ROIS_PER_BLOCK) ? (N - roiBase)
                                                     : ROIS_PER_BLOCK;
  if (cstart >= P || nroiLoc <= 0) return;
  const int npts = (cstart + CHUNK_PTS <= P) ? CHUNK_PTS : (P - cstart);

  // ---- 1) kick off the async coord DMA first (b128 body + b32 tail) ----
  const unsigned sp_base = lds_off32(&sp[0]);
  const float* gsrc = pts + (size_t)cstart * 3;
  const int ndw = npts * 3;
  const int nq  = ndw >> 2;
  for (int i = tid; i < nq; i += TPB) {
    ASYNC_LDS_B128(sp_base + ((unsigned)i << 4),
                   (unsigned long long)(size_t)(gsrc + (i << 2)));
  }
  const int rem = ndw & 3, base4 = ndw & ~3;
  if (tid < rem) {
    ASYNC_LDS_B32(sp_base + (unsigned)((base4 + tid) << 2),
                  (unsigned long long)(size_t)(gsrc + base4 + tid));
  }

  // ---- 2) overlap: derive roi constants while the DMA is in flight ----
  for (int r = tid; r < nroiLoc; r += TPB) {
    const float* q = rois + (size_t)(roiBase + r) * 7;
    float cx = q[0], cy = q[1], cz = q[2];
    float dx = q[3], dy = q[4], dz = q[5], rz = q[6];
    cz += 0.5f * dz;                       // bottom center -> box center
    float c = cosf(rz), s = sinf(rz);      // cos(-rz)=c, sin(-rz)=-s
    sroi[r][0] = cx;  sroi[r][1] = cy;  sroi[r][2] = cz;
    sroi[r][3] = 0.5f * dx;  sroi[r][4] = 0.5f * dy;  sroi[r][5] = 0.5f * dz;
    sroi[r][6] = c;   sroi[r][7] = s;
    sroi[r][8]  = (float)OUTG / dx;
    sroi[r][9]  = (float)OUTG / dy;
    sroi[r][10] = (float)OUTG / dz;
  }

  // ---- 3) one wait + one barrier covers both LDS structures ----
  WAIT_ASYNC0();
  __syncthreads();

  // ---- 4) pull this thread's points into registers ----
  float px[PTS_PER_THREAD], py[PTS_PER_THREAD], pz[PTS_PER_THREAD];
  int   pid[PTS_PER_THREAD];
  int np = 0;
  for (int j = tid; j < npts; j += TPB) {
    px[np] = sp[3 * j + 0];
    py[np] = sp[3 * j + 1];
    pz[np] = sp[3 * j + 2];
    pid[np] = cstart + j;
    ++np;
  }

  // ---- 5) sweep the roi group (LDS broadcast reads) ----
  for (int r = 0; r < nroiLoc; ++r) {
    const float cx = sroi[r][0], cy = sroi[r][1], cz = sroi[r][2];
    const float hx = sroi[r][3], hy = sroi[r][4], hz = sroi[r][5];
    const float c  = sroi[r][6], s  = sroi[r][7];
    const float ivx = sroi[r][8], ivy = sroi[r][9], ivz = sroi[r][10];
    const size_t rg = (size_t)(roiBase + r);
    for (int k = 0; k < np; ++k) {
      const float sx = px[k] - cx;
      const float sy = py[k] - cy;
      const float lz = pz[k] - cz;
      const float lx = sx * c + sy * s;    // rotate by -rz
      const float ly = sy * c - sx * s;
      if (fabsf(lz) <= hz && fabsf(lx) < hx && fabsf(ly) < hy) {
        int xi = (int)floorf((lx + hx) * ivx);
        int yi = (int)floorf((ly + hy) * ivy);
        int zi = (int)floorf((lz + hz) * ivz);
        xi = min(max(xi, 0), OUTG - 1);
        yi = min(max(yi, 0), OUTG - 1);
        zi = min(max(zi, 0), OUTG - 1);
        const int vid = (xi * OUTG + yi) * OUTG + zi;
        const size_t obase = (rg * VOXN + vid) * CCH;
        const float4* f4 = (const float4*)(feat + (size_t)pid[k] * CCH);
        if (mode == 0) {
#pragma unroll
          for (int q = 0; q < 4; ++q) {
            float4 v = f4[q];
            atomicMax(&out_u[obase + 4 * q + 0], enc_f32(v.x));
            atomicMax(&out_u[obase + 4 * q + 1], enc_f32(v.y));
            atomicMax(&out_u[obase + 4 * q + 2], enc_f32(v.z));
            atomicMax(&out_u[obase + 4 * q + 3], enc_f32(v.w));
          }
        } else if (has_cnt) {
          atomicAdd(&cnt[rg * VOXN + vid], 1u);
#pragma unroll
          for (int q = 0; q < 4; ++q) {
            float4 v = f4[q];
            atomicAdd(&out_f[obase + 4 * q + 0], v.x);
            atomicAdd(&out_f[obase + 4 * q + 1], v.y);
            atomicAdd(&out_f[obase + 4 * q + 2], v.z);
            atomicAdd(&out_f[obase + 4 * q + 3], v.w);
          }
        }
      }
    }
  }
}

// ---------------------------------------------------------------------------
__global__ void roiaware_finalize(unsigned* __restrict__ out_u,
                                  float* __restrict__ out_f,
                                  const unsigned* __restrict__ cnt,
                                  const int* __restrict__ mode_p,
                                  int total, int has_cnt) {
  const int mode = mode_p[0];
  for (int i = blockIdx.x * blockDim.x + threadIdx.x; i < total;
       i += gridDim.x * blockDim.x) {
    if (mode == 0) {
      const unsigned e = out_u[i];
      out_f[i] = (e == ENC_NEGINF) ? 0.0f : dec_f32(e);
    } else if (has_cnt) {
      const unsigned cv = cnt[i / CCH];
      out_f[i] = out_f[i] / (float)(cv > 1u ? cv : 1u);
    }
  }
}

// ---------------------------------------------------------------------------
extern "C" void kernel_launch(void* const* d_in, const int* in_sizes, int n_in,
                              void* d_out, int out_size, void* d_ws,
                              size_t ws_size, hipStream_t stream) {
  const float* rois = (const float*)d_in[0];
  const float* pts  = (const float*)d_in[1];
  const float* feat = (const float*)d_in[2];
  const int*   mode = (const int*)d_in[3];

  const int N = in_sizes[0] / 7;
  const int P = in_sizes[1] / 3;
  const int total = N * VOXN * CCH;    // == out_size
  const int nvox  = N * VOXN;

  unsigned* out_u = (unsigned*)d_out;
  float*    out_f = (float*)d_out;
  unsigned* cnt   = (unsigned*)d_ws;
  const int has_cnt = (ws_size >= (size_t)nvox * sizeof(unsigned)) ? 1 : 0;

  const int initBlocks = (total + TPB - 1) / TPB;
  roiaware_init<<<initBlocks, TPB, 0, stream>>>(out_u, cnt, mode, total, nvox,
                                                has_cnt);

  dim3 poolGrid((P + CHUNK_PTS - 1) / CHUNK_PTS,
                (N + ROIS_PER_BLOCK - 1) / ROIS_PER_BLOCK);
  roiaware_pool<<<poolGrid, TPB, 0, stream>>>(rois, pts, feat, mode, out_u,
                                              out_f, cnt, N, P, has_cnt);

  roiaware_finalize<<<initBlocks, TPB, 0, stream>>>(out_u, out_f, cnt, mode,
                                                    total, has_cnt);
}